// BehaviorFire_43748536877189
// MI455X (gfx1250) — compile-verified
//
#include <hip/hip_runtime.h>
#include <stdint.h>

// BehaviorFire (powderworld) fused stencil for gfx1250 (MI455X).
// Memory-bound op: no WMMA (wrong tool for 3x3 ones-kernel sums of one-hot
// data); instead wave32 LDS-staged fused stencil + async global->LDS loads
// (ASYNCcnt) + global_prefetch_b8.
//
// Inputs (setup_inputs order):
//  0 world [B,16,H,W] f32   1 rand_movement (unused)  2 rand_interact [B,1,H,W]
//  3 rand_element [B,1,H,W] 4 velocity_field [B,2,H,W] 5 did_gravity (unused)
//  6..13 constants (unused: kernel is ones, dir/onehot vecs are hardcoded)
// Outputs: w4 [B,16,H,W] then vf [B,2,H,W], concatenated flat.

#define BB    32
#define CC    16
#define HH    256
#define WW    256
#define HWSZ  (HH * WW)          // 65536
#define TILE  32
#define HALO  3
#define RR    (TILE + 2 * HALO)  // 38
#define RP    40                 // padded LDS row stride

// element ids
#define E_EMPTY 0
#define E_WATER 3
#define E_GAS   4
#define E_WOOD  5
#define E_ICE   6
#define E_FIRE  7
#define E_PLANT 8
#define E_LAVA  10
#define E_DUST  12

#if defined(__gfx1250__) && __has_builtin(__builtin_amdgcn_global_load_async_to_lds_b32)
#define USE_ASYNC_LDS 1
typedef __attribute__((address_space(1))) int* gas1_i32p;
typedef __attribute__((address_space(3))) int* gas3_i32p;
#else
#define USE_ASYNC_LDS 0
#endif

__global__ __launch_bounds__(256) void behavior_fire_tile_kernel(
    const float* __restrict__ world,         // [B,16,H,W]
    const float* __restrict__ rand_interact, // [B,1,H,W]
    const float* __restrict__ rand_element,  // [B,1,H,W]
    float* __restrict__ out_world,           // [B,16,H,W]
    float* __restrict__ s_out)               // [B,H,W] scratch: vf source term
{
  __shared__ float    s_fal[RR * RP];  // fire+lava (original world)
  __shared__ unsigned s_msk[RR * RP];  // packed per-pixel flags
  __shared__ float    s_s[RR * RP];    // 2*m + 20*md
  __shared__ unsigned s_m2[RR * RP];   // b0: burnables_w2, b1: does_burn, b2: does_burn_ice
  __shared__ float    s_g[RR * RP];    // hbn*fal + lava
  __shared__ float    s_hbn[RR * RP];  // conv3(burnables_w2)
#if USE_ASYNC_LDS
  __shared__ float    s_re[TILE * TILE];
#endif

  const int b   = blockIdx.z;
  const int tx0 = blockIdx.x * TILE;
  const int ty0 = blockIdx.y * TILE;
  const int tid = threadIdx.x;

  const float* wb  = world         + (size_t)b * CC * HWSZ;
  const float* bcb = rand_interact + (size_t)b * HWSZ;
  const float* reb = rand_element  + (size_t)b * HWSZ;
  float*       ob  = out_world     + (size_t)b * CC * HWSZ;

#if USE_ASYNC_LDS
  // Async-stage the rand_element tile into LDS (ASYNCcnt path) so the load
  // overlaps the 3 stencil stages. Each thread loads exactly the i's it
  // consumes in stage 4, so its own s_wait_asynccnt(0) is sufficient.
  for (int i = tid; i < TILE * TILE; i += 256) {
    const int py = i >> 5, px = i & 31;
    const size_t p = ((size_t)(ty0 + py) << 8) + (size_t)(tx0 + px);
    __builtin_amdgcn_global_load_async_to_lds_b32(
        (gas1_i32p)(reb + p),
        (gas3_i32p)&s_re[i],
        0, 0);
  }
#endif

  // Warm L2/WGP$ for the 16-channel copy in stage 4 (global_prefetch_b8).
  for (int i = tid; i < CC * TILE; i += 256) {
    const int c = i >> 5, row = i & 31;
    __builtin_prefetch(wb + ((size_t)c << 16) + ((size_t)(ty0 + row) << 8) + tx0, 0, 0);
  }

  // ---- Stage 1: load halo region (38x38): fal + packed flags -------------
  for (int i = tid; i < RR * RR; i += 256) {
    const int ly = i / RR, lx = i - ly * RR;
    const int gy = ty0 + ly - HALO, gx = tx0 + lx - HALO;
    float fal = 0.0f;
    unsigned m = 0u;
    if (gy >= 0 && gy < HH && gx >= 0 && gx < WW) {
      const size_t p = ((size_t)gy << 8) + (size_t)gx;
      const float fire  = wb[(size_t)E_FIRE  * HWSZ + p];
      const float lava  = wb[(size_t)E_LAVA  * HWSZ + p];
      const float wood  = wb[(size_t)E_WOOD  * HWSZ + p];
      const float plant = wb[(size_t)E_PLANT * HWSZ + p];
      const float gas   = wb[(size_t)E_GAS   * HWSZ + p];
      const float dust  = wb[(size_t)E_DUST  * HWSZ + p];
      const float ice   = wb[(size_t)E_ICE   * HWSZ + p];
      const float empt  = wb[(size_t)E_EMPTY * HWSZ + p];
      const float bc    = bcb[p];
      fal = fire + lava;
      const bool bW = wood > 0.5f, bP = plant > 0.5f, bG = gas > 0.5f;
      const bool bD = dust > 0.5f, bI = ice > 0.5f;
      const bool cand = (bW && (bc < 0.05f)) || ((bP || bG) && (bc < 0.2f)) || bD;
      const bool iceC = bI && (bc < 0.2f);
      const bool burnable = bW || bP || bG || bD;
      m = (unsigned)cand
        | ((unsigned)bD            << 1)
        | ((unsigned)iceC          << 2)
        | ((unsigned)burnable      << 3)
        | ((unsigned)(lava > 0.5f) << 4)
        | ((unsigned)(fire > 0.5f) << 5)
        | ((unsigned)(empt > 0.5f) << 6)
        | ((unsigned)(bc < 0.3f)   << 7);
    }
    s_fal[ly * RP + lx] = fal;
    s_msk[ly * RP + lx] = m;
  }
  __syncthreads();

  // ---- Stage 2: region [1,37): has_fire_neighbor -> burn flags, s --------
  for (int i = tid; i < RR * RR; i += 256) {
    const int ly = i / RR, lx = i - ly * RR;
    if (ly >= 1 && ly < RR - 1 && lx >= 1 && lx < RR - 1) {
      const int id = ly * RP + lx;
      const float n =
          s_fal[id - RP - 1] + s_fal[id - RP] + s_fal[id - RP + 1] +
          s_fal[id - 1]      + s_fal[id]      + s_fal[id + 1] +
          s_fal[id + RP - 1] + s_fal[id + RP] + s_fal[id + RP + 1];
      const bool hfn = n > 0.0f;
      const unsigned m = s_msk[id];
      const bool burn  = ((m & 1u) != 0u) && hfn;                  // does_burn
      const bool dustb = (((m >> 1) & 1u) != 0u) && hfn;           // dust burn
      const bool iceb  = (((m >> 2) & 1u) != 0u) && hfn;           // does_burn_ice
      const bool burn2 = (((m >> 3) & 1u) != 0u) && !burn;         // burnables of w2
      s_s[id]  = 2.0f * (float)burn + 20.0f * (float)dustb;
      s_m2[id] = (unsigned)burn2 | ((unsigned)burn << 1) | ((unsigned)iceb << 2);
    }
  }
  __syncthreads();

  // ---- Stage 3: region [2,36): hbn = conv3(burnables_w2), g --------------
  for (int i = tid; i < RR * RR; i += 256) {
    const int ly = i / RR, lx = i - ly * RR;
    if (ly >= 2 && ly < RR - 2 && lx >= 2 && lx < RR - 2) {
      const int id = ly * RP + lx;
      const unsigned c =
          (s_m2[id - RP - 1] & 1u) + (s_m2[id - RP] & 1u) + (s_m2[id - RP + 1] & 1u) +
          (s_m2[id - 1] & 1u)      + (s_m2[id] & 1u)      + (s_m2[id + 1] & 1u) +
          (s_m2[id + RP - 1] & 1u) + (s_m2[id + RP] & 1u) + (s_m2[id + RP + 1] & 1u);
      const float hbn  = (float)c;
      const float lava = ((s_msk[id] >> 4) & 1u) ? 1.0f : 0.0f;
      s_g[id]   = hbn * s_fal[id] + lava;   // fwbn + lava_w2
      s_hbn[id] = hbn;
    }
  }
#if USE_ASYNC_LDS
#if __has_builtin(__builtin_amdgcn_s_wait_asynccnt)
  __builtin_amdgcn_s_wait_asynccnt(0);
#else
  asm volatile("s_wait_asynccnt 0" ::: "memory");
#endif
#endif
  __syncthreads();

  // ---- Stage 4: interior 32x32: in_fire_range, final element, stores -----
  for (int i = tid; i < TILE * TILE; i += 256) {
    const int py = i >> 5, px = i & 31;
    const int ly = py + HALO, lx = px + HALO;
    const int id = ly * RP + lx;
    const float ifr =
        s_g[id - RP - 1] + s_g[id - RP] + s_g[id - RP + 1] +
        s_g[id - 1]      + s_g[id]      + s_g[id + 1] +
        s_g[id + RP - 1] + s_g[id + RP] + s_g[id + RP + 1];
    const unsigned m  = s_msk[id];
    const unsigned m2 = s_m2[id];
    const bool burn  = ((m2 >> 1) & 1u) != 0u;
    const bool iceb  = ((m2 >> 2) & 1u) != 0u;
    const bool empt  = ((m  >> 6) & 1u) != 0u;
    const bool fire0 = ((m  >> 5) & 1u) != 0u;
    const bool bc3   = ((m  >> 7) & 1u) != 0u;
    const bool dbe   = empt && (ifr > 0.0f) && bc3;      // does_burn_empty

    const size_t p = ((size_t)(ty0 + py) << 8) + (size_t)(tx0 + px);
#if USE_ASYNC_LDS
    const float re = s_re[i];
#else
    const float re = reb[p];
#endif
    const bool fire3 = fire0 || burn || dbe;
    const bool dfte  = fire3 && (re < 0.4f) && (s_hbn[id] == 0.0f);

    int oid = -1;
    if (dfte)             oid = E_EMPTY;
    else if (burn || dbe) oid = E_FIRE;
    else if (iceb)        oid = E_WATER;

    const float* wp = wb + p;
    float*       op = ob + p;
#pragma unroll
    for (int c = 0; c < CC; ++c) {
      float v = wp[(size_t)c << 16];
      if (oid >= 0) v = (c == oid) ? 1.0f : 0.0f;
      op[(size_t)c << 16] = v;
    }
    s_out[((size_t)b << 16) + p] = s_s[id];
  }
}

// Periodic (jnp.roll) velocity-field update from the spilled source term.
// s in {0, 2, 22}: m = (s>0), md = (s>20); apply the reference's exact
// two-step rounding order: vf -= 2*(...); vf -= 20*(...).
__global__ __launch_bounds__(256) void behavior_fire_vf_kernel(
    const float* __restrict__ vf_in,  // [B,2,H,W]
    const float* __restrict__ sbuf,   // [B,H,W]
    float* __restrict__ vf_out)       // [B,2,H,W]
{
  const int idx = blockIdx.x * 256 + threadIdx.x;  // < B*H*W
  const int b = idx >> 16;
  const int p = idx & 0xFFFF;
  const int y = p >> 8, x = p & 0xFF;

  const float* sb = sbuf + ((size_t)b << 16);
  const float sU = sb[(((y - 1) & 255) << 8) | x];
  const float sD = sb[(((y + 1) & 255) << 8) | x];
  const float sL = sb[(y << 8) | ((x - 1) & 255)];
  const float sR = sb[(y << 8) | ((x + 1) & 255)];

  const float mU = sU > 0.0f ? 1.0f : 0.0f, dU = sU > 20.0f ? 1.0f : 0.0f;
  const float mD = sD > 0.0f ? 1.0f : 0.0f, dD = sD > 20.0f ? 1.0f : 0.0f;
  const float mL = sL > 0.0f ? 1.0f : 0.0f, dL = sL > 20.0f ? 1.0f : 0.0f;
  const float mR = sR > 0.0f ? 1.0f : 0.0f, dR = sR > 20.0f ? 1.0f : 0.0f;

  const size_t base = ((size_t)b << 17) + (size_t)p;  // b*2*HW + p
  float vy = vf_in[base];
  float vx = vf_in[base + HWSZ];
  vy = vy - 2.0f  * (mD - mU);
  vy = vy - 20.0f * (dD - dU);
  vx = vx - 2.0f  * (mR - mL);
  vx = vx - 20.0f * (dR - dL);
  vf_out[base]        = vy;
  vf_out[base + HWSZ] = vx;
}

extern "C" void kernel_launch(void* const* d_in, const int* in_sizes, int n_in,
                              void* d_out, int out_size, void* d_ws, size_t ws_size,
                              hipStream_t stream) {
  (void)in_sizes; (void)n_in; (void)out_size; (void)ws_size;
  const float* world         = (const float*)d_in[0];
  const float* rand_interact = (const float*)d_in[2];
  const float* rand_element  = (const float*)d_in[3];
  const float* vf_in         = (const float*)d_in[4];

  float* out_world = (float*)d_out;
  float* out_vf    = (float*)d_out + (size_t)BB * CC * HWSZ;  // after w4
  float* sbuf      = (float*)d_ws;                            // B*H*W f32 = 8 MB

  dim3 grid1(WW / TILE, HH / TILE, BB);  // 8 x 8 x 32
  behavior_fire_tile_kernel<<<grid1, 256, 0, stream>>>(
      world, rand_interact, rand_element, out_world, sbuf);

  const int n = BB * HWSZ;  // 2,097,152
  behavior_fire_vf_kernel<<<n / 256, 256, 0, stream>>>(vf_in, sbuf, out_vf);
}